// AttentionPooling_33371895890590
// MI455X (gfx1250) — compile-verified
//
#include <hip/hip_runtime.h>
#include <hip/hip_bf16.h>

typedef __attribute__((ext_vector_type(2))) float v2f;
typedef __attribute__((ext_vector_type(8))) float v8f;

#define NROWS 1000000
#define DDIM  128
#define SEGS  50000

// Native non-returning f32 global atomic add (STOREcnt; drained at s_endpgm).
// Forces global_atomic_add_f32 instead of a possible CAS-loop expansion.
__device__ __forceinline__ void globalAtomicAddF32(float* p, float v) {
    asm volatile("global_atomic_add_f32 %0, %1, off" :: "v"(p), "v"(v) : "memory");
}

// ---------------------------------------------------------------------------
// Kernel 0: zero out / segmax / segsum (harness does not re-zero between runs)
// ---------------------------------------------------------------------------
__global__ void init_kernel(float* __restrict__ out,
                            unsigned* __restrict__ segmax,
                            float* __restrict__ segsum) {
    int i = blockIdx.x * blockDim.x + threadIdx.x;
    if (i < SEGS * DDIM) out[i] = 0.0f;
    if (i < SEGS) { segmax[i] = 0u; segsum[i] = 0.0f; }
}

// ---------------------------------------------------------------------------
// Kernel 1: gate = x @ Wg + bg ; segment max via monotone-encoded uint atomics
// One wave32 per row; float4 loads (coalesced 512B per row), shfl reduction.
// ---------------------------------------------------------------------------
__global__ void __launch_bounds__(256)
gate_kernel(const float* __restrict__ x, const int* __restrict__ index,
            const float* __restrict__ Wg, const float* __restrict__ bg,
            float* __restrict__ gate, unsigned* __restrict__ segmax) {
    const int wave = threadIdx.x >> 5;
    const int lane = threadIdx.x & 31;
    const int row  = blockIdx.x * 8 + wave;
    if (row >= NROWS) return;

    const float4 xv = *(const float4*)(x  + (size_t)row * DDIM + lane * 4);
    const float4 wv = *(const float4*)(Wg + lane * 4);
    float d = xv.x * wv.x + xv.y * wv.y + xv.z * wv.z + xv.w * wv.w;
    #pragma unroll
    for (int o = 16; o > 0; o >>= 1) d += __shfl_xor(d, o, 32);

    if (lane == 0) {
        float g = d + bg[0];
        gate[row] = g;
        unsigned u   = __float_as_uint(g);
        unsigned enc = (u & 0x80000000u) ? ~u : (u | 0x80000000u);
        atomicMax(&segmax[index[row]], enc);
    }
}

// ---------------------------------------------------------------------------
// Kernel 2: e = exp(gate - segmax[idx]); segsum[idx] += e (overwrite gate buf)
// ---------------------------------------------------------------------------
__global__ void __launch_bounds__(256)
softmax_kernel(const int* __restrict__ index, float* __restrict__ gate_e,
               const unsigned* __restrict__ segmax, float* __restrict__ segsum) {
    int i = blockIdx.x * blockDim.x + threadIdx.x;
    if (i >= NROWS) return;
    int s = index[i];
    unsigned m    = segmax[s];
    unsigned bits = (m & 0x80000000u) ? (m & 0x7FFFFFFFu) : ~m;
    float mx = __uint_as_float(bits);
    float e  = __expf(gate_e[i] - mx);
    gate_e[i] = e;
    globalAtomicAddF32(&segsum[s], e);
}

// ---------------------------------------------------------------------------
// Kernel 3: msg = x @ Wm + bm via V_WMMA_F32_16X16X4_F32, then gated scatter-add.
// Block = 128 threads (4 waves). Each wave owns a 16-row tile (64 rows/block).
// K tiled in slabs of 32. Wm slab is stored in LDS k-pair-interleaved:
//   lds_wm[(k>>1)*256 + n*2 + (k&1)] = Wm[k0+k][n]
// so each B fragment (K=kb,kb+1 at column n) is ONE aligned ds_load_b64 into a
// legal even-aligned WMMA operand pair -> no v_mov repacking churn.
// N = 1e6 is divisible by 64 -> EXEC is all ones around WMMA (required).
// ---------------------------------------------------------------------------
__global__ void __launch_bounds__(128)
pool_wmma_kernel(const float* __restrict__ x, const int* __restrict__ index,
                 const float* __restrict__ Wm, const float* __restrict__ bm,
                 const float* __restrict__ e_buf, const float* __restrict__ segsum,
                 float* __restrict__ out) {
    __shared__ float lds_wm[32 * DDIM];      // 16 KB K-slab of Wm (interleaved)
    __shared__ float lds_x[4][16 * 36];      // per-wave 16x32 x-tile, stride 36 pad
    __shared__ float lds_w[4][16];           // per-row softmax weight
    __shared__ int   lds_i[4][16];           // per-row segment id

    const int tid  = threadIdx.x;
    const int wave = tid >> 5;
    const int lane = tid & 31;
    const int r0   = blockIdx.x * 64 + wave * 16;

    // Stage per-row weight w = e / (segsum + 1e-10) and segment id.
    if (lane < 16) {
        int row = r0 + lane;
        int s   = index[row];
        lds_w[wave][lane] = e_buf[row] / (segsum[s] + 1e-10f);
        lds_i[wave][lane] = s;
    }

    v8f acc[8] = {};                         // 8 n-tiles of 16x16 f32 accumulators

    const int mrow = lane & 15;              // A-matrix M / B-matrix N for this lane
    const int koff = (lane >> 4) << 1;       // lanes 16-31 hold K+2,K+3 (koff even)

    for (int ks = 0; ks < 4; ++ks) {
        const int k0 = ks * 32;

        // Wm slab (contiguous 4096 floats in global) -> k-pair-interleaved LDS.
        {
            const float4* src = (const float4*)(Wm + (size_t)k0 * DDIM);
            #pragma unroll
            for (int t = 0; t < 8; ++t) {
                int    e4 = tid + t * 128;          // float4 index, 0..1023
                float4 v  = src[e4];
                int    k  = (e4 * 4) >> 7;          // slab row 0..31
                int    n  = (e4 * 4) & 127;         // col 0..124 (step 4)
                float* d  = &lds_wm[(k >> 1) * 256 + n * 2 + (k & 1)];
                d[0] = v.x; d[2] = v.y; d[4] = v.z; d[6] = v.w;
            }
        }
        // x tile slab: 16 rows x 32 cols, coalesced 128B per row-load.
        #pragma unroll
        for (int r = 0; r < 16; ++r)
            lds_x[wave][r * 36 + lane] = x[(size_t)(r0 + r) * DDIM + k0 + lane];
        __syncthreads();

        // Preload A fragments (aligned 8B LDS loads: kb even, stride 36 even).
        v2f a[8];
        #pragma unroll
        for (int kk = 0; kk < 8; ++kk) {
            int kb = kk * 4 + koff;
            a[kk] = *(const v2f*)&lds_x[wave][mrow * 36 + kb];
        }
        #pragma unroll
        for (int nt = 0; nt < 8; ++nt) {
            #pragma unroll
            for (int kk = 0; kk < 8; ++kk) {
                int kp = (kk * 4 + koff) >> 1;      // k-pair index (kb even)
                v2f b  = *(const v2f*)&lds_wm[kp * 256 + (nt * 16 + mrow) * 2];
                acc[nt] = __builtin_amdgcn_wmma_f32_16x16x4_f32(
                    false, a[kk], false, b, (short)0, acc[nt], false, false);
            }
        }
        __syncthreads();
    }

    // Epilogue: C/D layout -> VGPR j holds M=j (lanes 0-15) or M=j+8 (16-31).
    const int mbase = (lane >> 4) << 3;
    #pragma unroll
    for (int nt = 0; nt < 8; ++nt) {
        int   col = nt * 16 + mrow;
        float bmc = bm[col];
        #pragma unroll
        for (int j = 0; j < 8; ++j) {
            int   M   = j + mbase;
            float val = (acc[nt][j] + bmc) * lds_w[wave][M];
            globalAtomicAddF32(&out[(size_t)lds_i[wave][M] * DDIM + col], val);
        }
    }
}

// ---------------------------------------------------------------------------
extern "C" void kernel_launch(void* const* d_in, const int* in_sizes, int n_in,
                              void* d_out, int out_size, void* d_ws, size_t ws_size,
                              hipStream_t stream) {
    const float* x     = (const float*)d_in[0];
    const int*   index = (const int*)d_in[1];
    const float* Wg    = (const float*)d_in[2];
    const float* bg    = (const float*)d_in[3];
    const float* Wm    = (const float*)d_in[4];
    const float* bm    = (const float*)d_in[5];
    // d_in[6] = num_segments scalar; compile-time constant here.

    float*    out    = (float*)d_out;
    float*    gate   = (float*)d_ws;                    // N floats (becomes e)
    unsigned* segmax = (unsigned*)(gate + NROWS);       // S uints
    float*    segsum = (float*)(segmax + SEGS);         // S floats

    {   // init: covers SEGS*DDIM = 6.4M elements
        int total = SEGS * DDIM;
        init_kernel<<<(total + 255) / 256, 256, 0, stream>>>(out, segmax, segsum);
    }
    gate_kernel<<<NROWS / 8, 256, 0, stream>>>(x, index, Wg, bg, gate, segmax);
    softmax_kernel<<<(NROWS + 255) / 256, 256, 0, stream>>>(index, gate, segmax, segsum);
    pool_wmma_kernel<<<NROWS / 64, 128, 0, stream>>>(x, index, Wm, bm, gate, segsum, out);
}